// SmallEncoder_75282186764688
// MI455X (gfx1250) — compile-verified
//
#include <hip/hip_runtime.h>

#define DEG 6
#define GROUPS 8

typedef float     v2f  __attribute__((ext_vector_type(2)));
typedef float     v8f  __attribute__((ext_vector_type(8)));

// Branchless guarded load: address clamp (v_cndmask) + value mask (v_cndmask).
// No EXEC divergence, so waves stay fully active around WMMA.
__device__ __forceinline__ float ldmask(const float* __restrict__ p, long idx,
                                        bool ok, long clampIdx) {
  float v = p[ok ? idx : clampIdx];
  return ok ? v : 0.f;
}

// -------------------------------------------------------------------------
// Lx: out[b,n,c] = alpha * sum_{j<6} ew[n*6+j] * in[b, src[n*6+j], c]
//                  (+ beta * prev[b,n,c])
// dst = repeat(arange(N),6) in the reference graph, so scatter-add == gather.
// Scalar variant (C==3) and float4 variant (C%4==0).
// -------------------------------------------------------------------------
__global__ __launch_bounds__(256) void lx_kernel(
    const float* __restrict__ in, const float* __restrict__ prev,
    const int* __restrict__ src, const float* __restrict__ ew,
    float* __restrict__ out, int B, int N, int C, float alpha, float beta) {
  long idx = (long)blockIdx.x * blockDim.x + threadIdx.x;
  long total = (long)B * N * C;
  if (idx >= total) return;
  int c = (int)(idx % C);
  long bn = idx / C;
  int n = (int)(bn % N);
  int b = (int)(bn / N);
  const float* base = in + (long)b * N * C;
  int e0 = n * DEG;
  float s = 0.f;
#pragma unroll
  for (int j = 0; j < DEG; ++j) {
    int sn = src[e0 + j];
    s = fmaf(ew[e0 + j], base[(long)sn * C + c], s);
  }
  float r = alpha * s;
  if (beta != 0.f) r = fmaf(beta, prev[idx], r);
  out[idx] = r;
}

__global__ __launch_bounds__(256) void lx4_kernel(
    const float4* __restrict__ in, const float4* __restrict__ prev,
    const int* __restrict__ src, const float* __restrict__ ew,
    float4* __restrict__ out, int B, int N, int C4, float alpha, float beta) {
  long idx = (long)blockIdx.x * blockDim.x + threadIdx.x;
  long total = (long)B * N * C4;
  if (idx >= total) return;
  int c = (int)(idx % C4);
  long bn = idx / C4;
  int n = (int)(bn % N);
  int b = (int)(bn / N);
  const float4* base = in + (long)b * N * C4;
  int e0 = n * DEG;
  float sx = 0.f, sy = 0.f, sz = 0.f, sw = 0.f;
#pragma unroll
  for (int j = 0; j < DEG; ++j) {
    int sn = src[e0 + j];
    float w = ew[e0 + j];
    float4 v = base[(long)sn * C4 + c];
    sx = fmaf(w, v.x, sx); sy = fmaf(w, v.y, sy);
    sz = fmaf(w, v.z, sz); sw = fmaf(w, v.w, sw);
  }
  float4 r;
  r.x = alpha * sx; r.y = alpha * sy; r.z = alpha * sz; r.w = alpha * sw;
  if (beta != 0.f) {
    float4 p = prev[idx];
    r.x = fmaf(beta, p.x, r.x); r.y = fmaf(beta, p.y, r.y);
    r.z = fmaf(beta, p.z, r.z); r.w = fmaf(beta, p.w, r.w);
  }
  out[idx] = r;
}

// -------------------------------------------------------------------------
// Chebyshev channel mix via WMMA, templated on (CIN, COUT, K):
//   out[m, o] = bias[o] + sum_k sum_c T_k[m, c] * W[k, c, o]
// One wave owns a 64x16 output supertile (4 row tiles): the weight (B)
// fragment is loaded once per k-step and reused by 4 independent WMMAs.
// A fragments are b64 loads (contiguous pair) on the unguarded path.
// fp32 path: V_WMMA_F32_16X16X4_F32
//   A 16x4:  lanes 0-15 hold K=0,1 (a.x,a.y), lanes 16-31 K=2,3
//   B 4x16:  lanes 0-15 hold K=0,1, lanes 16-31 K=2,3; N = lane&15
//   C/D 16x16: VGPR i -> M = i + 8*(lane>=16), N = lane&15
// -------------------------------------------------------------------------
template <int CIN, int COUT, int K>
__global__ __launch_bounds__(128) void cheb_mix_t(
    const float* __restrict__ t0, const float* __restrict__ t1,
    const float* __restrict__ t2, const float* __restrict__ W,
    const float* __restrict__ bias, float* __restrict__ out, int M) {
  constexpr int RT = 4;                       // row tiles per wave
  constexpr int tilesN = (COUT + 15) / 16;
  constexpr bool colGuard = (COUT % 16) != 0;
  constexpr bool kGuard   = (CIN % 4) != 0;

  int lane = threadIdx.x & 31;
  int wave = blockIdx.x * (blockDim.x >> 5) + (threadIdx.x >> 5);
  int tilesM = M >> 6;                        // 64-row supertiles (M % 64 == 0)
  if (wave >= tilesM * tilesN) return;        // uniform per wave
  int tm = wave % tilesM, tn = wave / tilesM;
  int row0 = tm << 6, col0 = tn << 4;
  int r = lane & 15, hi = lane >> 4;
  int col = col0 + r;
  bool cok = !colGuard || (col < COUT);
  long colc = colGuard ? (cok ? col : (COUT - 1)) : col;   // clamped index

  v8f acc[RT];
#pragma unroll
  for (int t = 0; t < RT; ++t) acc[t] = v8f{0.f,0.f,0.f,0.f,0.f,0.f,0.f,0.f};

  const float* Ts[3] = {t0, t1, t2};
#pragma unroll
  for (int k = 0; k < K; ++k) {
    const float* __restrict__ T  = Ts[k];
    const float* __restrict__ Wk = W + (long)k * CIN * COUT;
#pragma unroll
    for (int cb = 0; cb < CIN; cb += 4) {
      int ka = cb + 2 * hi;
      v2f bf;
      if constexpr (kGuard || colGuard) {
        long k0 = (ka     < CIN) ? ka     : 0;
        long k1 = (ka + 1 < CIN) ? ka + 1 : 0;
        bf.x = ldmask(Wk, k0 * COUT + colc, cok && (ka     < CIN), 0);
        bf.y = ldmask(Wk, k1 * COUT + colc, cok && (ka + 1 < CIN), 0);
      } else {
        bf.x = Wk[(long)ka * COUT + col];
        bf.y = Wk[(long)(ka + 1) * COUT + col];
      }
#pragma unroll
      for (int t = 0; t < RT; ++t) {
        const float* __restrict__ Ar = T + (long)(row0 + t * 16 + r) * CIN;
        v2f a;
        if constexpr (kGuard) {
          a.x = ldmask(Ar, ka,     ka     < CIN, 0);
          a.y = ldmask(Ar, ka + 1, ka + 1 < CIN, 0);
        } else {
          a = *reinterpret_cast<const v2f*>(Ar + ka);   // b64 load
        }
        acc[t] = __builtin_amdgcn_wmma_f32_16x16x4_f32(
            false, a, false, bf, (short)0, acc[t], false, false);
      }
    }
  }

  if (cok) {
    float bv = bias[col];
#pragma unroll
    for (int t = 0; t < RT; ++t) {
#pragma unroll
      for (int i = 0; i < 8; ++i) {
        int row = row0 + t * 16 + i + hi * 8;
        out[(long)row * COUT + col] = acc[t][i] + bv;
      }
    }
  }
}

// -------------------------------------------------------------------------
// GroupNorm, deterministic two-stage reduction over (N, C/G) per (b,g)
// -------------------------------------------------------------------------
__global__ __launch_bounds__(256) void gn_stats1(
    const float* __restrict__ x, float* __restrict__ partial,
    int B, int N, int C, int nchunk) {
  int gc = C / GROUPS;
  int bg = blockIdx.y;
  int b = bg / GROUPS, g = bg % GROUPS;
  long per = ((long)N + nchunk - 1) / nchunk;
  long n0 = (long)blockIdx.x * per;
  long n1 = n0 + per; if (n1 > N) n1 = N;
  float s = 0.f, s2 = 0.f;
  long cnt = (n1 > n0) ? (n1 - n0) * gc : 0;
  for (long i = threadIdx.x; i < cnt; i += blockDim.x) {
    long n = n0 + i / gc;
    int  j = (int)(i % gc);
    float v = x[((long)b * N + n) * C + g * gc + j];
    s += v; s2 += v * v;
  }
  __shared__ float sh0[256];
  __shared__ float sh1[256];
  sh0[threadIdx.x] = s; sh1[threadIdx.x] = s2;
  __syncthreads();
  for (int off = 128; off > 0; off >>= 1) {
    if ((int)threadIdx.x < off) {
      sh0[threadIdx.x] += sh0[threadIdx.x + off];
      sh1[threadIdx.x] += sh1[threadIdx.x + off];
    }
    __syncthreads();
  }
  if (threadIdx.x == 0) {
    long o = ((long)bg * nchunk + blockIdx.x) * 2;
    partial[o] = sh0[0]; partial[o + 1] = sh1[0];
  }
}

// float4 stats when gc % 4 == 0 (C = 32, 64, 96); GC4 = gc/4 compile-time.
template <int GC4>
__global__ __launch_bounds__(256) void gn_stats1_v4(
    const float4* __restrict__ x, float* __restrict__ partial,
    int B, int N, int C4, int nchunk) {
  int bg = blockIdx.y;
  int b = bg / GROUPS, g = bg % GROUPS;
  long per = ((long)N + nchunk - 1) / nchunk;
  long n0 = (long)blockIdx.x * per;
  long n1 = n0 + per; if (n1 > N) n1 = N;
  float s = 0.f, s2 = 0.f;
  long cnt = (n1 > n0) ? (n1 - n0) * GC4 : 0;
  for (long i = threadIdx.x; i < cnt; i += blockDim.x) {
    long n = n0 + i / GC4;
    int  j = (int)(i % GC4);
    float4 v = x[((long)b * N + n) * C4 + g * GC4 + j];
    s  += v.x + v.y + v.z + v.w;
    s2 += v.x * v.x + v.y * v.y + v.z * v.z + v.w * v.w;
  }
  __shared__ float sh0[256];
  __shared__ float sh1[256];
  sh0[threadIdx.x] = s; sh1[threadIdx.x] = s2;
  __syncthreads();
  for (int off = 128; off > 0; off >>= 1) {
    if ((int)threadIdx.x < off) {
      sh0[threadIdx.x] += sh0[threadIdx.x + off];
      sh1[threadIdx.x] += sh1[threadIdx.x + off];
    }
    __syncthreads();
  }
  if (threadIdx.x == 0) {
    long o = ((long)bg * nchunk + blockIdx.x) * 2;
    partial[o] = sh0[0]; partial[o + 1] = sh1[0];
  }
}

__global__ __launch_bounds__(64) void gn_stats2(
    const float* __restrict__ partial, float* __restrict__ stats,
    int nchunk, float count) {
  int bg = blockIdx.x;
  __shared__ float sh0[64];
  __shared__ float sh1[64];
  float s = 0.f, s2 = 0.f;
  for (int i = threadIdx.x; i < nchunk; i += blockDim.x) {
    s  += partial[((long)bg * nchunk + i) * 2];
    s2 += partial[((long)bg * nchunk + i) * 2 + 1];
  }
  sh0[threadIdx.x] = s; sh1[threadIdx.x] = s2;
  __syncthreads();
  for (int off = 32; off > 0; off >>= 1) {
    if ((int)threadIdx.x < off) {
      sh0[threadIdx.x] += sh0[threadIdx.x + off];
      sh1[threadIdx.x] += sh1[threadIdx.x + off];
    }
    __syncthreads();
  }
  if (threadIdx.x == 0) {
    float mean = sh0[0] / count;
    float var  = sh1[0] / count - mean * mean;
    if (var < 0.f) var = 0.f;
    stats[bg * 2]     = mean;
    stats[bg * 2 + 1] = rsqrtf(var + 1e-5f);
  }
}

// Vectorized apply; GC = channels per group (compile-time -> div is mul/shift).
// Optionally fuses residual add + relu (the tail of each bottleneck).
template <int GC>
__global__ __launch_bounds__(256) void gn_apply4(
    const float4* __restrict__ x, const float* __restrict__ stats,
    const float4* __restrict__ gamma, const float4* __restrict__ beta,
    const float4* __restrict__ residual, float4* __restrict__ out,
    int B, int N, int C4, int relu) {
  long idx = (long)blockIdx.x * blockDim.x + threadIdx.x;
  long total = (long)B * N * C4;
  if (idx >= total) return;
  int c4 = (int)(idx % C4);
  int b  = (int)(idx / ((long)N * C4));
  float4 v  = x[idx];
  float4 gm = gamma[c4];
  float4 bt = beta[c4];
  int cb = c4 * 4;
  float vi[4] = {v.x, v.y, v.z, v.w};
  float gi[4] = {gm.x, gm.y, gm.z, gm.w};
  float bi[4] = {bt.x, bt.y, bt.z, bt.w};
  float oo[4];
#pragma unroll
  for (int i = 0; i < 4; ++i) {
    int g  = (cb + i) / GC;
    int bg = b * GROUPS + g;
    oo[i] = (vi[i] - stats[bg * 2]) * stats[bg * 2 + 1] * gi[i] + bi[i];
  }
  if (residual) {
    float4 rv = residual[idx];
    oo[0] += rv.x; oo[1] += rv.y; oo[2] += rv.z; oo[3] += rv.w;
  }
  if (relu) {
#pragma unroll
    for (int i = 0; i < 4; ++i) oo[i] = fmaxf(oo[i], 0.f);
  }
  float4 o; o.x = oo[0]; o.y = oo[1]; o.z = oo[2]; o.w = oo[3];
  out[idx] = o;
}

__global__ __launch_bounds__(256) void pool4v_kernel(
    const float4* __restrict__ x, float4* __restrict__ out,
    int B, int No, int C4) {   // No = N/4
  long idx = (long)blockIdx.x * blockDim.x + threadIdx.x;
  long total = (long)B * No * C4;
  if (idx >= total) return;
  int c = (int)(idx % C4);
  long bn = idx / C4;
  int n = (int)(bn % No);
  int b = (int)(bn / No);
  const float4* p = x + (((long)b * No * 4 + (long)n * 4) * C4 + c);
  float4 a0 = p[0], a1 = p[C4], a2 = p[2 * (long)C4], a3 = p[3 * (long)C4];
  float4 o;
  o.x = fmaxf(fmaxf(a0.x, a1.x), fmaxf(a2.x, a3.x));
  o.y = fmaxf(fmaxf(a0.y, a1.y), fmaxf(a2.y, a3.y));
  o.z = fmaxf(fmaxf(a0.z, a1.z), fmaxf(a2.z, a3.z));
  o.w = fmaxf(fmaxf(a0.w, a1.w), fmaxf(a2.w, a3.w));
  out[idx] = o;
}

// -------------------------------------------------------------------------
// Host-side orchestration
// -------------------------------------------------------------------------
static const int  Bn = 4, N0v = 49152, N1v = 12288, N2v = 3072;
static const long SL_BIG = (long)Bn * N0v * 64;   // 12,582,912 floats
static const long SL_MID = (long)Bn * N0v * 16;   //  3,145,728 floats
static const int  NCHUNK = 64;

static inline void launch_lx(hipStream_t s, const float* in, const float* prev,
                             const int* src, const float* ew, float* out,
                             int B, int N, int C, float alpha, float beta) {
  int thr = 256;
  if ((C & 3) == 0) {
    int C4 = C >> 2;
    long total = (long)B * N * C4;
    lx4_kernel<<<(unsigned)((total + thr - 1) / thr), thr, 0, s>>>(
        (const float4*)in, (const float4*)prev, src, ew, (float4*)out,
        B, N, C4, alpha, beta);
  } else {
    long total = (long)B * N * C;
    lx_kernel<<<(unsigned)((total + thr - 1) / thr), thr, 0, s>>>(
        in, prev, src, ew, out, B, N, C, alpha, beta);
  }
}

static void launch_mix(hipStream_t s, const float* t0, const float* t1,
                       const float* t2, const float* W, const float* bias,
                       float* out, int M, int Cin, int Cout, int K) {
  const float* t1c = t1 ? t1 : t0;
  const float* t2c = t2 ? t2 : t0;
  int tilesM = M >> 6;
  int tilesN = (Cout + 15) >> 4;
  long waves = (long)tilesM * tilesN;
  int wpb = 4, thr = 32 * wpb;
  unsigned blocks = (unsigned)((waves + wpb - 1) / wpb);
#define MIX_CASE(CI, CO, KK)                                                   \
  if (Cin == CI && Cout == CO && K == KK) {                                    \
    cheb_mix_t<CI, CO, KK><<<blocks, thr, 0, s>>>(t0, t1c, t2c, W, bias, out, M); \
    return;                                                                    \
  }
  MIX_CASE(3, 32, 3)
  MIX_CASE(32, 16, 3)
  MIX_CASE(16, 16, 3)
  MIX_CASE(16, 64, 3)
  MIX_CASE(32, 64, 1)
  MIX_CASE(64, 16, 3)
  MIX_CASE(64, 24, 3)
  MIX_CASE(24, 24, 3)
  MIX_CASE(24, 96, 3)
  MIX_CASE(64, 96, 1)
  MIX_CASE(96, 24, 3)
  MIX_CASE(96, 128, 3)
#undef MIX_CASE
}

static void cheb_run(hipStream_t s, const float* X, const int* src, const float* ew,
                     const float* W, const float* bias, int K,
                     int B, int N, int Cin, int Cout,
                     float* T1, float* T2, float* OUT) {
  const float* t1 = nullptr;
  const float* t2 = nullptr;
  if (K >= 2) { launch_lx(s, X,  nullptr, src, ew, T1, B, N, Cin, 1.f,  0.f); t1 = T1; }
  if (K >= 3) { launch_lx(s, T1, X,       src, ew, T2, B, N, Cin, 2.f, -1.f); t2 = T2; }
  launch_mix(s, X, t1, t2, W, bias, OUT, B * N, Cin, Cout, K);
}

static void gn_run(hipStream_t s, const float* X, const float* gamma, const float* beta,
                   const float* residual, int relu, float* out,
                   int B, int N, int C, float* part, float* stats) {
  int gc = C / GROUPS;
  dim3 g1(NCHUNK, B * GROUPS);
  if ((gc & 3) == 0) {
    int C4 = C >> 2;
    switch (gc >> 2) {
      case 1: gn_stats1_v4<1><<<g1, 256, 0, s>>>((const float4*)X, part, B, N, C4, NCHUNK); break;
      case 2: gn_stats1_v4<2><<<g1, 256, 0, s>>>((const float4*)X, part, B, N, C4, NCHUNK); break;
      case 3: gn_stats1_v4<3><<<g1, 256, 0, s>>>((const float4*)X, part, B, N, C4, NCHUNK); break;
      default: gn_stats1<<<g1, 256, 0, s>>>(X, part, B, N, C, NCHUNK); break;
    }
  } else {
    gn_stats1<<<g1, 256, 0, s>>>(X, part, B, N, C, NCHUNK);
  }
  gn_stats2<<<Bn * GROUPS, 64, 0, s>>>(part, stats, NCHUNK, (float)((long)N * gc));
  long total4 = (long)B * N * (C >> 2);
  int thr = 256;
  unsigned blocks = (unsigned)((total4 + thr - 1) / thr);
  const float4* x4 = (const float4*)X;
  const float4* g4 = (const float4*)gamma;
  const float4* b4 = (const float4*)beta;
  const float4* r4 = (const float4*)residual;
  float4*       o4 = (float4*)out;
  int C4 = C >> 2;
  switch (gc) {
    case 2:  gn_apply4<2> <<<blocks, thr, 0, s>>>(x4, stats, g4, b4, r4, o4, B, N, C4, relu); break;
    case 3:  gn_apply4<3> <<<blocks, thr, 0, s>>>(x4, stats, g4, b4, r4, o4, B, N, C4, relu); break;
    case 4:  gn_apply4<4> <<<blocks, thr, 0, s>>>(x4, stats, g4, b4, r4, o4, B, N, C4, relu); break;
    case 8:  gn_apply4<8> <<<blocks, thr, 0, s>>>(x4, stats, g4, b4, r4, o4, B, N, C4, relu); break;
    case 12: gn_apply4<12><<<blocks, thr, 0, s>>>(x4, stats, g4, b4, r4, o4, B, N, C4, relu); break;
    default: gn_apply4<4> <<<blocks, thr, 0, s>>>(x4, stats, g4, b4, r4, o4, B, N, C4, relu); break;
  }
}

// Weight leaf order: pytree (sorted-key) flatten of the weights dict:
// blocks[0..3] then conv1.{W,b}, conv2.{W,b}, norm1.{b,g}.
// Per block: c1.W,c1.b,c2.W,c2.b,c3.W,c3.b,n1.b,n1.g,n2.b,n2.g,n3.b,n3.g[,ns.b,ns.g,sc.W,sc.b]
static void bottleneck_run(hipStream_t s, void* const* d_in, int base,
                           const float* X, const int* src, const float* ew,
                           int B, int N, int Cin, int Cout, bool has_sc,
                           float* T1, float* T2, float* Y1, float* Y2,
                           float* Y3, float* SCb, float* part, float* stats,
                           float* OUT) {
  auto Wp = [&](int i) { return (const float*)d_in[7 + base + i]; };
  int mid = Cout / 4;
  cheb_run(s, X,  src, ew, Wp(0), Wp(1), 3, B, N, Cin, mid,  T1, T2, Y1);
  gn_run (s, Y1, Wp(7),  Wp(6),  nullptr, 1, Y1, B, N, mid,  part, stats);
  cheb_run(s, Y1, src, ew, Wp(2), Wp(3), 3, B, N, mid, mid,  T1, T2, Y2);
  gn_run (s, Y2, Wp(9),  Wp(8),  nullptr, 1, Y2, B, N, mid,  part, stats);
  cheb_run(s, Y2, src, ew, Wp(4), Wp(5), 3, B, N, mid, Cout, T1, T2, Y3);
  const float* res = X;
  if (has_sc) {
    cheb_run(s, X, src, ew, Wp(14), Wp(15), 1, B, N, Cin, Cout, T1, T2, SCb);
    gn_run (s, SCb, Wp(13), Wp(12), nullptr, 0, SCb, B, N, Cout, part, stats);
    res = SCb;
  }
  // Fused: OUT = relu(groupnorm(Y3) + res)
  gn_run(s, Y3, Wp(11), Wp(10), res, 1, OUT, B, N, Cout, part, stats);
}

extern "C" void kernel_launch(void* const* d_in, const int* in_sizes, int n_in,
                              void* d_out, int out_size, void* d_ws, size_t ws_size,
                              hipStream_t stream) {
  (void)in_sizes; (void)n_in; (void)out_size; (void)ws_size;
  const float* x   = (const float*)d_in[0];
  const int*   ei0 = (const int*)d_in[1];   // [2,E]; first E entries are src
  const float* ew0 = (const float*)d_in[2];
  const int*   ei1 = (const int*)d_in[3];
  const float* ew1 = (const float*)d_in[4];
  const int*   ei2 = (const int*)d_in[5];
  const float* ew2 = (const float*)d_in[6];
  auto Wl = [&](int i) { return (const float*)d_in[7 + i]; };

  float* ws = (float*)d_ws;
  float* H0    = ws;
  float* H1    = ws + SL_BIG;
  float* T1b   = ws + 2 * SL_BIG;
  float* T2b   = ws + 3 * SL_BIG;
  float* Y3b   = ws + 4 * SL_BIG;
  float* SCb   = ws + 5 * SL_BIG;
  float* Y1b   = ws + 6 * SL_BIG;
  float* Y2b   = ws + 6 * SL_BIG + SL_MID;
  float* part  = ws + 6 * SL_BIG + 2 * SL_MID;          // 32*64*2 floats
  float* stats = part + (long)Bn * GROUPS * NCHUNK * 2; // 64 floats

  // conv1 (3->32, K=3) + norm1 + relu, at N0
  cheb_run(stream, x, ei0, ew0, Wl(56), Wl(57), 3, Bn, N0v, 3, 32, T1b, T2b, H0);
  gn_run (stream, H0, Wl(61), Wl(60), nullptr, 1, H0, Bn, N0v, 32, part, stats);

  // block0: 32->64 (sc) at N0
  bottleneck_run(stream, d_in, 0, H0, ei0, ew0, Bn, N0v, 32, 64, true,
                 T1b, T2b, Y1b, Y2b, Y3b, SCb, part, stats, H1);
  // block1: 64->64 at N0
  bottleneck_run(stream, d_in, 16, H1, ei0, ew0, Bn, N0v, 64, 64, false,
                 T1b, T2b, Y1b, Y2b, Y3b, SCb, part, stats, H0);
  // pool N0 -> N1 (C=64)
  {
    long total = (long)Bn * N1v * 16; int thr = 256;
    pool4v_kernel<<<(unsigned)((total + thr - 1) / thr), thr, 0, stream>>>(
        (const float4*)H0, (float4*)H1, Bn, N1v, 16);
  }
  // block2: 64->96 (sc) at N1
  bottleneck_run(stream, d_in, 28, H1, ei1, ew1, Bn, N1v, 64, 96, true,
                 T1b, T2b, Y1b, Y2b, Y3b, SCb, part, stats, H0);
  // block3: 96->96 at N1
  bottleneck_run(stream, d_in, 44, H0, ei1, ew1, Bn, N1v, 96, 96, false,
                 T1b, T2b, Y1b, Y2b, Y3b, SCb, part, stats, H1);
  // pool N1 -> N2 (C=96)
  {
    long total = (long)Bn * N2v * 24; int thr = 256;
    pool4v_kernel<<<(unsigned)((total + thr - 1) / thr), thr, 0, stream>>>(
        (const float4*)H1, (float4*)H0, Bn, N2v, 24);
  }
  // conv2 (96->128, K=3) at N2 -> d_out
  cheb_run(stream, H0, ei2, ew2, Wl(58), Wl(59), 3, Bn, N2v, 96, 128,
           T1b, T2b, (float*)d_out);
}